// CovariantPotentialNet_44135083933998
// MI455X (gfx1250) — compile-verified
//
#include <hip/hip_runtime.h>
#include <math.h>

// MI455X / gfx1250: wave32. All matrix ops use V_WMMA_F32_16X16X4_F32 (exact
// fp32 path — after algebraic collapse the whole net is ~200 MFLOP vs ~4.5 MB
// of HBM traffic, i.e. bandwidth/latency bound, so the fp32 matrix pipe is
// free and preserves reference numerics at fp32 precision).

typedef __attribute__((ext_vector_type(2))) float v2f;
typedef __attribute__((ext_vector_type(8))) float v8f;

#define B_N  4096
#define D_N  64
#define K_N  64
#define DM_N 512

// workspace layout (float offsets) — all 8-byte aligned
#define WS_SIG    0        // [2]  sigma_z, sigma_v
#define WS_CONSTS 2        // [2]  h0, cconst
#define WS_CQ     16       // [512]
#define WS_WEFF   528      // [512]
#define WS_H      1040     // [64]
#define WS_G0     1104     // [64]
#define WS_CVEFF  1168     // [64]
#define WS_CN2    1232     // [64]
#define WS_DENC   1296     // [64]
#define WS_CEK    2048     // [64*512]
#define WS_CEV    34816    // [64*512]
#define WS_AMAT   67584    // [512*64]
#define WS_W2     100352   // [128*64]  rows 0-63 = G, rows 64-127 = projected centers
#define WS_P      108544   // [4096*128]

// ---------------------------------------------------------------------------
// f32 WMMA GEMM: C[M,N] = scale * A[M,K] @ (TRANSB ? B[N,K]^T : B[K,N])
// One wave computes a 16 x (16*NB) strip: one A fragment is reused across NB
// independent WMMA accumulator chains per K-step. Fragments for step k+4 are
// prefetched into registers before the WMMAs of step k issue (software
// pipelining), so loads overlap the matrix pipe instead of stalling it.
// Wave32 lane mapping per CDNA5 ISA 7.12.2 (V_WMMA_F32_16X16X4_F32):
//   A frag: lane L -> M = L%16, element v -> K = k0 + 2*(L/16) + v
//   B frag: lane L -> N = L%16, element v -> K = k0 + 2*(L/16) + v
//   C/D   : vgpr r -> M = r + 8*(L/16), N = L%16
// ---------------------------------------------------------------------------
template <int TRANSB, int SCALE_RECIP, int NB>
__global__ void wmma_gemm_f32(const float* __restrict__ A,
                              const float* __restrict__ Bm,
                              float* __restrict__ C,
                              int M, int N, int K,
                              const float* __restrict__ scale_ptr)
{
    const int lane    = threadIdx.x & 31;
    const int wave    = threadIdx.x >> 5;
    const int groupsN = (N >> 4) / NB;
    const int grp     = blockIdx.x * (blockDim.x >> 5) + wave;
    if (grp >= (M >> 4) * groupsN) return;
    const int tm  = grp / groupsN;
    const int tg  = grp - tm * groupsN;
    const int n0  = tg * NB * 16;           // first column of this wave's strip
    const int m   = lane & 15;
    const int hi  = lane >> 4;

    float s = 1.0f;
    if (SCALE_RECIP) s = 1.0f / scale_ptr[0];

    v8f acc[NB];
#pragma unroll
    for (int t = 0; t < NB; ++t) acc[t] = (v8f){};

    const float* arow = A + (size_t)(tm * 16 + m) * K + 2 * hi;
    const float* brow = Bm + (size_t)(n0 + m) * K + 2 * hi;   // TRANSB path

    // --- prologue: load fragments for k0 = 0 ---------------------------------
    v2f va = *(const v2f*)(arow);
    v2f vb[NB];
#pragma unroll
    for (int t = 0; t < NB; ++t) {
        if (TRANSB) {
            vb[t] = *(const v2f*)(brow + (size_t)t * 16 * K);
        } else {
            const int kk = 2 * hi;
            vb[t][0] = Bm[(size_t)kk * N + n0 + t * 16 + m];
            vb[t][1] = Bm[(size_t)(kk + 1) * N + n0 + t * 16 + m];
        }
    }

    // --- steady state: prefetch k0+4, compute k0 -----------------------------
    for (int k0 = 0; k0 < K - 4; k0 += 4) {
        const int k1 = k0 + 4;
        v2f va_n = *(const v2f*)(arow + k1);
        v2f vb_n[NB];
#pragma unroll
        for (int t = 0; t < NB; ++t) {
            if (TRANSB) {
                vb_n[t] = *(const v2f*)(brow + (size_t)t * 16 * K + k1);
            } else {
                const int kk = k1 + 2 * hi;
                vb_n[t][0] = Bm[(size_t)kk * N + n0 + t * 16 + m];
                vb_n[t][1] = Bm[(size_t)(kk + 1) * N + n0 + t * 16 + m];
            }
        }
#pragma unroll
        for (int t = 0; t < NB; ++t)
            acc[t] = __builtin_amdgcn_wmma_f32_16x16x4_f32(
                false, va, false, vb[t], (short)0, acc[t], false, false);
        va = va_n;
#pragma unroll
        for (int t = 0; t < NB; ++t) vb[t] = vb_n[t];
    }
    // --- epilogue step -------------------------------------------------------
#pragma unroll
    for (int t = 0; t < NB; ++t)
        acc[t] = __builtin_amdgcn_wmma_f32_16x16x4_f32(
            false, va, false, vb[t], (short)0, acc[t], false, false);

#pragma unroll
    for (int t = 0; t < NB; ++t) {
#pragma unroll
        for (int r = 0; r < 8; ++r) {
            const int row = tm * 16 + r + 8 * hi;
            const int col = n0 + t * 16 + m;
            C[(size_t)row * N + col] = acc[t][r] * s;
        }
    }
}

// ---------------------------------------------------------------------------
// Spectral norms: sigma_z of zW [512,64] (5 power iterations, torch
// spectral_norm semantics) and sigma_v = ||vW|| (exact for a 1xDM matrix).
// One block of 64 threads (2 waves).
// ---------------------------------------------------------------------------
__global__ void spectral_kernel(const float* __restrict__ zW,
                                const float* __restrict__ vW,
                                float* __restrict__ sig)
{
    __shared__ float u[DM_N];
    __shared__ float v[D_N];
    __shared__ float red[D_N];
    __shared__ float snorm;
    const int t = threadIdx.x;              // 0..63

    const float u0 = 1.0f / (sqrtf((float)DM_N) + 1e-12f);
    for (int i = t; i < DM_N; i += D_N) u[i] = u0;
    __syncthreads();

    float last_n2 = 0.0f;
    for (int iter = 0; iter < 5; ++iter) {
        // v = l2n(W^T u): thread t owns column t
        float a = 0.0f;
        for (int i = 0; i < DM_N; ++i) a += zW[i * D_N + t] * u[i];
        v[t] = a;
        red[t] = a * a;
        __syncthreads();
        for (int off = 32; off > 0; off >>= 1) {
            if (t < off) red[t] += red[t + off];
            __syncthreads();
        }
        if (t == 0) snorm = sqrtf(red[0]) + 1e-12f;
        __syncthreads();
        v[t] = v[t] / snorm;
        __syncthreads();
        // u = l2n(W v): thread t owns rows t, t+64, ...
        float tmp[8]; float ss = 0.0f;
#pragma unroll
        for (int r = 0; r < 8; ++r) {
            const int i = t + D_N * r;
            float b = 0.0f;
            for (int j = 0; j < D_N; ++j) b += zW[i * D_N + j] * v[j];
            tmp[r] = b; ss += b * b;
        }
        red[t] = ss;
        __syncthreads();
        for (int off = 32; off > 0; off >>= 1) {
            if (t < off) red[t] += red[t + off];
            __syncthreads();
        }
        if (t == 0) snorm = red[0];
        __syncthreads();
        last_n2 = snorm;
        const float un = sqrtf(last_n2) + 1e-12f;
#pragma unroll
        for (int r = 0; r < 8; ++r) u[t + D_N * r] = tmp[r] / un;
        __syncthreads();
    }
    // sigma = u . (W v) with u = (Wv)/(||Wv||+eps)  ->  n2/(sqrt(n2)+eps)
    if (t == 0) sig[0] = last_n2 / (sqrtf(last_n2) + 1e-12f);

    float sv = 0.0f;
    for (int i = t; i < DM_N; i += D_N) { const float x = vW[i]; sv += x * x; }
    red[t] = sv;
    __syncthreads();
    for (int off = 32; off > 0; off >>= 1) {
        if (t < off) red[t] += red[t + off];
        __syncthreads();
    }
    if (t == 0) sig[1] = sqrtf(red[0]);
}

// ---------------------------------------------------------------------------
// Small-vector precompute (one block, 512 threads). Needs CEk/CEv/Amat ready.
//   c_q    = Wq @ zb + bq
//   w_eff  = (vW/sigma_v) @ Wo
//   h      = A^T @ bk,   h0 = c_q . bk
//   g0[k]  = CEk[k] . c_q,  cv_eff[k] = CEv[k] . w_eff
//   cconst = bv.w_eff + bo.(vW/sigma_v) + vb
//   projected centers -> W2 rows 64..127, plus |c|^2 and (1-|c|^2)
// ---------------------------------------------------------------------------
__global__ void precompute_small(const float* __restrict__ Wq,
                                 const float* __restrict__ bq,
                                 const float* __restrict__ zb,
                                 const float* __restrict__ bk,
                                 const float* __restrict__ vW,
                                 const float* __restrict__ vb,
                                 const float* __restrict__ Wo,
                                 const float* __restrict__ bo,
                                 const float* __restrict__ bv,
                                 const float* __restrict__ centers,
                                 float* __restrict__ ws)
{
    const int t = threadIdx.x;              // 0..511
    const float* sig  = ws + WS_SIG;
    float* c_q   = ws + WS_CQ;
    float* w_eff = ws + WS_WEFF;
    const float* CEk  = ws + WS_CEK;
    const float* CEv  = ws + WS_CEV;
    const float* Amat = ws + WS_AMAT;
    float* h    = ws + WS_H;
    float* g0   = ws + WS_G0;
    float* cve  = ws + WS_CVEFF;
    float* cn2  = ws + WS_CN2;
    float* denc = ws + WS_DENC;
    float* W2   = ws + WS_W2;

    const float invsv = 1.0f / sig[1];

    float a = bq[t];
    const float* wqr = Wq + (size_t)t * DM_N;
    for (int f = 0; f < DM_N; ++f) a += wqr[f] * zb[f];
    c_q[t] = a;

    float w = 0.0f;
    for (int e = 0; e < DM_N; ++e) w += vW[e] * Wo[(size_t)e * DM_N + t];
    w_eff[t] = w * invsv;
    __syncthreads();

    if (t < K_N) {
        float hh = 0.0f;
        for (int i = 0; i < DM_N; ++i) hh += Amat[(size_t)i * D_N + t] * bk[i];
        h[t] = hh;

        float gg = 0.0f, cc = 0.0f;
        const float* cek = CEk + (size_t)t * DM_N;
        const float* cev = CEv + (size_t)t * DM_N;
        for (int i = 0; i < DM_N; ++i) { gg += cek[i] * c_q[i]; cc += cev[i] * w_eff[i]; }
        g0[t]  = gg;
        cve[t] = cc;

        // project chart center t onto the open unit ball
        const float* cr = centers + (size_t)t * D_N;
        float n2 = 0.0f;
        for (int j = 0; j < D_N; ++j) n2 += cr[j] * cr[j];
        const float n = sqrtf(n2);
        const float scale = fminf(1.0f, (1.0f - 1e-5f) / fmaxf(n, 1e-12f));
        float pn2 = 0.0f;
        for (int j = 0; j < D_N; ++j) {
            const float pc = cr[j] * scale;
            W2[(size_t)(K_N + t) * D_N + j] = pc;
            pn2 += pc * pc;
        }
        cn2[t]  = pn2;
        denc[t] = 1.0f - pn2;
    }
    if (t == 0) {
        float h0 = 0.0f;
        for (int i = 0; i < DM_N; ++i) h0 += c_q[i] * bk[i];
        float cconst = vb[0];
        for (int i = 0; i < DM_N; ++i) cconst += bv[i] * w_eff[i] + bo[i] * vW[i] * invsv;
        ws[WS_CONSTS + 0] = h0;
        ws[WS_CONSTS + 1] = cconst;
    }
}

// ---------------------------------------------------------------------------
// Per-row epilogue: Poincare bias + online softmax + weighted value collapse.
// One thread per batch row.
// ---------------------------------------------------------------------------
__global__ void epilogue_kernel(const float* __restrict__ P,
                                const float* __restrict__ z,
                                const float* __restrict__ rw,
                                const float* __restrict__ geo_in,
                                const float* __restrict__ ws,
                                float* __restrict__ out)
{
    const int b = blockIdx.x * blockDim.x + threadIdx.x;
    if (b >= B_N) return;

    const float* g0   = ws + WS_G0;
    const float* cn2  = ws + WS_CN2;
    const float* denc = ws + WS_DENC;
    const float* cve  = ws + WS_CVEFF;
    const float* h    = ws + WS_H;
    const float h0     = ws[WS_CONSTS + 0];
    const float cconst = ws[WS_CONSTS + 1];
    const float geo    = geo_in[0];
    const float invs   = 0.04419417382415922f;  // 1/sqrt(512)

    // zn2 = |z_b|^2, zh = z_b . h
    const float* zp = z + (size_t)b * D_N;
    float zn2 = 0.0f, zh = 0.0f;
    for (int j = 0; j < D_N; ++j) { const float zj = zp[j]; zn2 += zj * zj; zh += zj * h[j]; }
    const float den1 = 1.0f - zn2;

    const float* Pr  = P  + (size_t)b * 128;
    const float* rwp = rw + (size_t)b * K_N;

    // online softmax with fused weighted numerator
    float mx = -3.402823466e38f, ssum = 0.0f, num = 0.0f;
    for (int k = 0; k < K_N; ++k) {
        const float s1   = Pr[k];          // z . G[k]
        const float zdot = Pr[64 + k];     // z . c_proj[k]
        const float r    = rwp[k];

        const float den = fmaxf(den1 * denc[k], 1e-12f);
        const float diff2 = zn2 - 2.0f * zdot + cn2[k];
        float arg = 1.0f + 2.0f * diff2 / den;
        arg = fmaxf(arg, 1.0f + 1e-7f);
        const float d  = acoshf(arg);
        const float sc = (r * (s1 + g0[k]) + zh + h0) * invs - geo * d * d;
        const float wv = r * cve[k];

        if (sc > mx) {
            const float c = __expf(mx - sc);
            ssum = ssum * c + 1.0f;
            num  = num  * c + wv;
            mx = sc;
        } else {
            const float e = __expf(sc - mx);
            ssum += e;
            num  += e * wv;
        }
    }
    out[b] = num / ssum + cconst;
}

// ---------------------------------------------------------------------------
template <int TRANSB, int SCALE_RECIP, int NB>
static inline void launch_gemm(const float* A, const float* Bm, float* C,
                               int M, int N, int K,
                               const float* scale_ptr, hipStream_t stream)
{
    const int waves  = (M / 16) * (N / 16) / NB;   // one wave per 16 x (16*NB) strip
    const int blocks = (waves + 7) / 8;            // 8 waves (256 threads) / block
    wmma_gemm_f32<TRANSB, SCALE_RECIP, NB>
        <<<blocks, 256, 0, stream>>>(A, Bm, C, M, N, K, scale_ptr);
}

extern "C" void kernel_launch(void* const* d_in, const int* in_sizes, int n_in,
                              void* d_out, int out_size, void* d_ws, size_t ws_size,
                              hipStream_t stream)
{
    (void)in_sizes; (void)n_in; (void)out_size; (void)ws_size;
    const float* z     = (const float*)d_in[0];
    const float* rw    = (const float*)d_in[1];
    const float* cemb  = (const float*)d_in[2];
    const float* ccent = (const float*)d_in[3];
    const float* zW    = (const float*)d_in[4];
    const float* zb    = (const float*)d_in[5];
    const float* Wq    = (const float*)d_in[6];
    const float* bq    = (const float*)d_in[7];
    const float* Wk    = (const float*)d_in[8];
    const float* bk    = (const float*)d_in[9];
    const float* Wv    = (const float*)d_in[10];
    const float* bv    = (const float*)d_in[11];
    const float* Wo    = (const float*)d_in[12];
    const float* bo    = (const float*)d_in[13];
    const float* geo   = (const float*)d_in[14];
    const float* vW    = (const float*)d_in[15];
    const float* vb    = (const float*)d_in[16];
    float* out = (float*)d_out;
    float* ws  = (float*)d_ws;

    // 1) spectral norms (sigma_z, sigma_v)
    spectral_kernel<<<1, 64, 0, stream>>>(zW, vW, ws + WS_SIG);
    // 2) CE_k = chart_emb @ Wk^T   [64 x 512]
    launch_gemm<1, 0, 4>(cemb, Wk, ws + WS_CEK, K_N, DM_N, DM_N, ws + WS_SIG, stream);
    // 3) CE_v = chart_emb @ Wv^T   [64 x 512]
    launch_gemm<1, 0, 4>(cemb, Wv, ws + WS_CEV, K_N, DM_N, DM_N, ws + WS_SIG, stream);
    // 4) A = (1/sigma_z) * Wq @ zW [512 x 64]
    launch_gemm<0, 1, 4>(Wq, zW, ws + WS_AMAT, DM_N, D_N, DM_N, ws + WS_SIG, stream);
    // 5) small vectors (c_q, w_eff, h, g0, cv_eff, consts, projected centers)
    precompute_small<<<1, 512, 0, stream>>>(Wq, bq, zb, bk, vW, vb, Wo, bo, bv,
                                            ccent, ws);
    // 6) G = CE_k @ A  ->  W2 rows 0..63   [64 x 64]
    launch_gemm<0, 0, 4>(ws + WS_CEK, ws + WS_AMAT, ws + WS_W2, K_N, D_N, DM_N,
                         ws + WS_SIG, stream);
    // 7) P = z @ W2^T   [4096 x 128]  (cols 0-63: z.G[k], cols 64-127: z.c[k])
    launch_gemm<1, 0, 4>(z, ws + WS_W2, ws + WS_P, B_N, 128, D_N,
                         ws + WS_SIG, stream);
    // 8) per-row softmax/value epilogue
    epilogue_kernel<<<B_N / 256, 256, 0, stream>>>(ws + WS_P, z, rw, geo, ws, out);
}